// BiasGatLayer_12146167513762
// MI455X (gfx1250) — compile-verified
//
#include <hip/hip_runtime.h>

#define LEAKY(v) ((v) >= 0.0f ? (v) : 0.1f * (v))

typedef __attribute__((ext_vector_type(2))) float v2f;
typedef __attribute__((ext_vector_type(8))) float v8f;

__device__ __forceinline__ v8f wmma_f32(v2f a, v2f b, v8f c) {
  // D = A(16x4) * B(4x16) + C, fp32 exact path on CDNA5
  return __builtin_amdgcn_wmma_f32_16x16x4_f32(false, a, false, b, (short)0, c, false, false);
}

__device__ __forceinline__ v2f lds_pair(const float* p) {
  return *reinterpret_cast<const v2f*>(p);  // 8B-aligned by construction -> ds_load_b64
}

__device__ __forceinline__ float wave_max32(float v) {
#pragma unroll
  for (int m = 16; m >= 1; m >>= 1) v = fmaxf(v, __shfl_xor(v, m, 32));
  return v;
}
__device__ __forceinline__ float wave_sum32(float v) {
#pragma unroll
  for (int m = 16; m >= 1; m >>= 1) v += __shfl_xor(v, m, 32);
  return v;
}

// ---------------------------------------------------------------------------
// K0: fold tiny embedding tables against their 32-vectors: 20 + 10 scalars.
__global__ void prep_tables_kernel(const float* __restrict__ path_table,
                                   const float* __restrict__ pconv_w,
                                   const float* __restrict__ edge_table,
                                   const float* __restrict__ econv_w,
                                   float* __restrict__ path_dot,
                                   float* __restrict__ edge_dot) {
  int t = threadIdx.x;
  if (t < 20) {
    float s = 0.0f;
#pragma unroll
    for (int d = 0; d < 32; ++d) s += path_table[t * 32 + d] * pconv_w[d];
    path_dot[t] = s;
  } else if (t >= 32 && t < 42) {
    int u = t - 32;
    float s = 0.0f;
#pragma unroll
    for (int d = 0; d < 32; ++d) s += edge_table[u * 32 + d] * econv_w[d];
    edge_dot[u] = s;
  }
}

// K0b: conv1_w [o][384] -> w1t [c][128]
__global__ void transpose_w1_kernel(const float* __restrict__ w1, float* __restrict__ w1t) {
  int idx = blockIdx.x * 256 + threadIdx.x;
  if (idx >= 128 * 384) return;
  int o = idx / 384, c = idx % 384;
  w1t[c * 128 + o] = w1[idx];
}

// K0c: conv2_w [o][c][3][3] -> w2t [di*3+dj][c][o]
__global__ void prep_w2_kernel(const float* __restrict__ w2, float* __restrict__ w2t) {
  int idx = blockIdx.x * 256 + threadIdx.x;
  if (idx >= 147456) return;
  int dj = idx % 3;
  int di = (idx / 3) % 3;
  int c = (idx / 9) % 128;
  int o = idx / 1152;
  w2t[((size_t)(di * 3 + dj) * 128 + c) * 128 + o] = w2[idx];
}

// ---------------------------------------------------------------------------
// K1: O[b] = W(128x128) @ X[b](128x64).  B operand pair-interleaved:
// XP[kp*160 + col*2 + (k&1)], kp=k/2; stride 160 = 32 mod 64 -> conflict-free b64.
__global__ __launch_bounds__(256) void gemm_wx_kernel(const float* __restrict__ W,
                                                      const float* __restrict__ X,
                                                      float* __restrict__ O) {
  extern __shared__ float smem[];
  float* XP = smem;            // 64 kpairs * 160
  float* Ws = smem + 10240;    // [128][129]
  int b = blockIdx.x;
  int tid = threadIdx.x, wave = tid >> 5, lane = tid & 31;
  const float* xg = X + (size_t)b * 8192;
  for (int t = tid; t < 8192; t += 256)
    XP[(t >> 7) * 160 + (t & 63) * 2 + ((t >> 6) & 1)] = xg[t];
  for (int t = tid; t < 16384; t += 256) Ws[(t >> 7) * 129 + (t & 127)] = W[t];
  __syncthreads();
  int m0 = wave * 16, hi = lane >> 4, lo = lane & 15;
  float* og = O + (size_t)b * 8192;
  v8f acc[4];
#pragma unroll
  for (int nt = 0; nt < 4; ++nt)
#pragma unroll
    for (int r = 0; r < 8; ++r) acc[nt][r] = 0.0f;
  for (int k0 = 0; k0 < 128; k0 += 4) {
    int kh = k0 + (hi << 1);
    v2f a; a.x = Ws[(m0 + lo) * 129 + kh]; a.y = Ws[(m0 + lo) * 129 + kh + 1];
    const float* bp = XP + (kh >> 1) * 160 + lo * 2;
#pragma unroll
    for (int nt = 0; nt < 4; ++nt) {
      v2f bb = lds_pair(bp + nt * 32);
      acc[nt] = wmma_f32(a, bb, acc[nt]);
    }
  }
#pragma unroll
  for (int nt = 0; nt < 4; ++nt) {
    int col = nt * 16 + lo;
#pragma unroll
    for (int r = 0; r < 8; ++r) og[(size_t)(m0 + r + 8 * hi) * 64 + col] = acc[nt][r];
  }
}

// K2: x_self / y_other (1024 dots of length 128)
__global__ void self_kernel(const float* __restrict__ xf, const float* __restrict__ yf,
                            const float* __restrict__ xc2_w, const float* __restrict__ xc2_b,
                            const float* __restrict__ yc2_w, const float* __restrict__ yc2_b,
                            float* __restrict__ xself, float* __restrict__ yoth) {
  int idx = blockIdx.x * 256 + threadIdx.x;
  if (idx < 512) {
    int b = idx >> 6, m = idx & 63;
    float s = 0.0f;
    const float* p = xf + (size_t)b * 8192 + m;
    for (int c = 0; c < 128; ++c) s += xc2_w[c] * p[c * 64];
    xself[idx] = s + xc2_b[0];
  } else if (idx < 1024) {
    int j = idx - 512;
    int b = j >> 6, n = j & 63;
    float s = 0.0f;
    const float* p = yf + (size_t)b * 8192 + n;
    for (int c = 0; c < 128; ++c) s += yc2_w[c] * p[c * 64];
    yoth[j] = s + yc2_b[0];
  }
}

// K3: L[b,i,j] = leaky(x_self[b,j] + y_other[b,i])
__global__ void logist_kernel(const float* __restrict__ xself, const float* __restrict__ yoth,
                              float* __restrict__ L) {
  int idx = blockIdx.x * 256 + threadIdx.x;  // 32768
  int b = idx >> 12, i = (idx >> 6) & 63, j = idx & 63;
  float v = xself[b * 64 + j] + yoth[b * 64 + i];
  L[idx] = LEAKY(v);
}

// K4: local[b,e,m,n] = L[b,m,n] + leaky(path_dot[path_mat]) + p_bias
__global__ void local_kernel(const float* __restrict__ L, const float* __restrict__ path_dot,
                             const int* __restrict__ path_mat, const float* __restrict__ p_bias,
                             float* __restrict__ locbuf) {
  size_t idx = (size_t)blockIdx.x * 256 + threadIdx.x;  // 2097152
  int n = (int)(idx & 63);
  int m = (int)((idx >> 6) & 63);
  int b = (int)(idx >> 18);
  float pl = path_dot[path_mat[idx]];
  pl = LEAKY(pl);
  locbuf[idx] = L[(size_t)b * 4096 + m * 64 + n] + pl + p_bias[idx];
}

// ---------------------------------------------------------------------------
// K5: per (b,e): both softmaxes of local[b,e] in LDS, then
//   x_ret[b,:,e,:] = leaky(y_f[b](CxN) @ A^T + x_f[b])
//   y_ret[b,:,e,:] = leaky(x_f[b](CxM) @ A2^T + y_f[b])
// Softmax matrices at ld=66 (even) so B-pair reads are 8B aligned.
__global__ __launch_bounds__(256) void attn_ret_kernel(const float* __restrict__ locg,
                                                       const float* __restrict__ xf,
                                                       const float* __restrict__ yf,
                                                       float* __restrict__ xret,
                                                       float* __restrict__ yret) {
  extern __shared__ float smem[];
  float* Aa  = smem;               // [64][66]  row softmax (over n)
  float* At  = smem + 4224;        // [64][66]  col softmax, stored [n][m]
  float* xfs = smem + 8448;        // [128][65]
  float* yfs = smem + 16768;       // [128][65]
  float* loc = smem + 25088;       // [64][65]
  int blk = blockIdx.x, b = blk >> 6, e = blk & 63;
  int tid = threadIdx.x, wave = tid >> 5, lane = tid & 31;
  const float* lg = locg + (size_t)(b * 64 + e) * 4096;
  for (int t = tid; t < 4096; t += 256) loc[(t >> 6) * 65 + (t & 63)] = lg[t];
  const float* xfg = xf + (size_t)b * 8192;
  const float* yfg = yf + (size_t)b * 8192;
  for (int t = tid; t < 8192; t += 256) {
    xfs[(t >> 6) * 65 + (t & 63)] = xfg[t];
    yfs[(t >> 6) * 65 + (t & 63)] = yfg[t];
  }
  __syncthreads();
  for (int r = wave; r < 64; r += 8) {
    float v0 = loc[r * 65 + lane], v1 = loc[r * 65 + lane + 32];
    float mx = wave_max32(fmaxf(v0, v1));
    float e0 = __expf(v0 - mx), e1 = __expf(v1 - mx);
    float inv = 1.0f / wave_sum32(e0 + e1);
    Aa[r * 66 + lane] = e0 * inv; Aa[r * 66 + lane + 32] = e1 * inv;
  }
  for (int n = wave; n < 64; n += 8) {
    float v0 = loc[lane * 65 + n], v1 = loc[(lane + 32) * 65 + n];
    float mx = wave_max32(fmaxf(v0, v1));
    float e0 = __expf(v0 - mx), e1 = __expf(v1 - mx);
    float inv = 1.0f / wave_sum32(e0 + e1);
    At[n * 66 + lane] = e0 * inv; At[n * 66 + lane + 32] = e1 * inv;
  }
  __syncthreads();
  int m0 = wave * 16, hi = lane >> 4, lo = lane & 15;
  float* xout = xret + (size_t)b * 128 * 4096 + e * 64;  // + c*4096 + m
  float* yout = yret + (size_t)b * 128 * 4096 + e * 64;  // + c*4096 + n
  {
    v8f acc[4];
#pragma unroll
    for (int nt = 0; nt < 4; ++nt)
#pragma unroll
      for (int r = 0; r < 8; ++r) acc[nt][r] = xfs[(m0 + r + 8 * hi) * 65 + nt * 16 + lo];
    for (int k0 = 0; k0 < 64; k0 += 4) {
      int kh = k0 + (hi << 1);
      v2f a; a.x = yfs[(m0 + lo) * 65 + kh]; a.y = yfs[(m0 + lo) * 65 + kh + 1];
#pragma unroll
      for (int nt = 0; nt < 4; ++nt) {
        v2f bb = lds_pair(Aa + (nt * 16 + lo) * 66 + kh);
        acc[nt] = wmma_f32(a, bb, acc[nt]);
      }
    }
#pragma unroll
    for (int nt = 0; nt < 4; ++nt) {
      int col = nt * 16 + lo;
#pragma unroll
      for (int r = 0; r < 8; ++r) {
        float v = acc[nt][r];
        xout[(size_t)(m0 + r + 8 * hi) * 4096 + col] = LEAKY(v);
      }
    }
  }
  {
    v8f acc[4];
#pragma unroll
    for (int nt = 0; nt < 4; ++nt)
#pragma unroll
      for (int r = 0; r < 8; ++r) acc[nt][r] = yfs[(m0 + r + 8 * hi) * 65 + nt * 16 + lo];
    for (int k0 = 0; k0 < 64; k0 += 4) {
      int kh = k0 + (hi << 1);
      v2f a; a.x = xfs[(m0 + lo) * 65 + kh]; a.y = xfs[(m0 + lo) * 65 + kh + 1];
#pragma unroll
      for (int nt = 0; nt < 4; ++nt) {
        v2f bb = lds_pair(At + (nt * 16 + lo) * 66 + kh);
        acc[nt] = wmma_f32(a, bb, acc[nt]);
      }
    }
#pragma unroll
    for (int nt = 0; nt < 4; ++nt) {
      int col = nt * 16 + lo;
#pragma unroll
      for (int r = 0; r < 8; ++r) {
        float v = acc[nt][r];
        yout[(size_t)(m0 + r + 8 * hi) * 4096 + col] = LEAKY(v);
      }
    }
  }
}

// ---------------------------------------------------------------------------
// K6: conv1 (1x1, 384->128) as GEMM. B pair-interleaved, stride 288.
__global__ __launch_bounds__(256) void conv1_kernel(const float* __restrict__ xret,
                                                    const float* __restrict__ yret,
                                                    const float* __restrict__ pre,
                                                    const float* __restrict__ w1t,
                                                    const float* __restrict__ bias,
                                                    float* __restrict__ hout) {
  extern __shared__ float smem[];
  float* BP = smem;          // 32 kpairs * 288 (128 cols * 2)
  float* As = smem + 9216;   // [128][65]
  int blk = blockIdx.x, b = blk >> 5, pc = (blk & 31) * 128;
  int tid = threadIdx.x, wave = tid >> 5, lane = tid & 31;
  int m0 = wave * 16, hi = lane >> 4, lo = lane & 15;
  v8f acc[8];
#pragma unroll
  for (int nt = 0; nt < 8; ++nt) {
#pragma unroll
    for (int r = 0; r < 8; ++r) acc[nt][r] = bias[m0 + r + 8 * hi];
  }
  for (int cb = 0; cb < 6; ++cb) {
    int c0 = cb * 64;
    __syncthreads();
    for (int t = tid; t < 8192; t += 256) {
      int cc = t >> 7, o = t & 127;
      As[o * 65 + cc] = w1t[(size_t)(c0 + cc) * 128 + o];
    }
    for (int t = tid; t < 8192; t += 256) {
      int kk = t >> 7, pp = t & 127;
      int c = c0 + kk;
      const float* src;
      if (c < 128)      src = xret + (size_t)(b * 128 + c) * 4096;
      else if (c < 256) src = yret + (size_t)(b * 128 + (c - 128)) * 4096;
      else              src = pre  + (size_t)(b * 128 + (c - 256)) * 4096;
      BP[(kk >> 1) * 288 + pp * 2 + (kk & 1)] = src[pc + pp];
    }
    __syncthreads();
    for (int k0 = 0; k0 < 64; k0 += 4) {
      int kh = k0 + (hi << 1);
      v2f a; a.x = As[(m0 + lo) * 65 + kh]; a.y = As[(m0 + lo) * 65 + kh + 1];
      const float* bp = BP + (kh >> 1) * 288 + lo * 2;
#pragma unroll
      for (int nt = 0; nt < 8; ++nt) {
        v2f bb = lds_pair(bp + nt * 32);
        acc[nt] = wmma_f32(a, bb, acc[nt]);
      }
    }
  }
  float* dst = hout + (size_t)b * 128 * 4096 + pc;
#pragma unroll
  for (int nt = 0; nt < 8; ++nt) {
    int col = nt * 16 + lo;
#pragma unroll
    for (int r = 0; r < 8; ++r) {
      float v = acc[nt][r];
      dst[(size_t)(m0 + r + 8 * hi) * 4096 + col] = LEAKY(v);
    }
  }
}

// K7: conv2 (3x3 SAME) as implicit GEMM. h rows pair-interleaved over c.
__global__ __launch_bounds__(256) void conv2_kernel(const float* __restrict__ h,
                                                    const float* __restrict__ w2t,
                                                    const float* __restrict__ bias,
                                                    float* __restrict__ out) {
  extern __shared__ float smem[];
  float* HP = smem;            // [3][64 cpairs * 160]
  float* As = smem + 30720;    // [128][65]
  int blk = blockIdx.x, b = blk >> 6, i = blk & 63;
  int tid = threadIdx.x, wave = tid >> 5, lane = tid & 31;
  for (int dd = 0; dd < 3; ++dd) {
    int ii = i - 1 + dd;
    if (ii >= 0 && ii < 64) {
      const float* src = h + (size_t)b * 128 * 4096 + ii * 64;
      for (int t = tid; t < 8192; t += 256) {
        int c = t >> 6, j = t & 63;
        HP[dd * 10240 + (c >> 1) * 160 + j * 2 + (c & 1)] = src[(size_t)c * 4096 + j];
      }
    } else {
      for (int t = tid; t < 8192; t += 256) {
        int c = t >> 6, j = t & 63;
        HP[dd * 10240 + (c >> 1) * 160 + j * 2 + (c & 1)] = 0.0f;
      }
    }
  }
  int m0 = wave * 16, hi = lane >> 4, lo = lane & 15;
  v8f acc[4];
#pragma unroll
  for (int nt = 0; nt < 4; ++nt) {
#pragma unroll
    for (int r = 0; r < 8; ++r) acc[nt][r] = bias[m0 + r + 8 * hi];
  }
  for (int dd = 0; dd < 3; ++dd) {
    for (int dj = 0; dj < 3; ++dj) {
      for (int kb = 0; kb < 2; ++kb) {
        __syncthreads();
        const float* wsrc = w2t + (size_t)(dd * 3 + dj) * 128 * 128;
        for (int t = tid; t < 8192; t += 256) {
          int cc = t >> 7, o = t & 127;
          As[o * 65 + cc] = wsrc[(size_t)(kb * 64 + cc) * 128 + o];
        }
        __syncthreads();
        const float* hrow = HP + dd * 10240 + kb * 32 * 160;  // cpair base
        for (int k0 = 0; k0 < 64; k0 += 4) {
          int kh = k0 + (hi << 1);
          v2f a; a.x = As[(m0 + lo) * 65 + kh]; a.y = As[(m0 + lo) * 65 + kh + 1];
          const float* bp = hrow + (kh >> 1) * 160;
#pragma unroll
          for (int nt = 0; nt < 4; ++nt) {
            int col = nt * 16 + lo;
            int jj = col + dj - 1;
            int jc = jj < 0 ? 0 : (jj > 63 ? 63 : jj);
            float ok = ((unsigned)jj < 64u) ? 1.0f : 0.0f;
            v2f bb = lds_pair(bp + jc * 2);
            bb.x *= ok; bb.y *= ok;
            acc[nt] = wmma_f32(a, bb, acc[nt]);
          }
        }
      }
    }
  }
  float* dst = out + (size_t)b * 128 * 4096 + i * 64;
#pragma unroll
  for (int nt = 0; nt < 4; ++nt) {
    int col = nt * 16 + lo;
#pragma unroll
    for (int r = 0; r < 8; ++r) {
      float v = acc[nt][r];
      dst[(size_t)(m0 + r + 8 * hi) * 4096 + col] = LEAKY(v);
    }
  }
}

// K8: g = L + leaky(score.men2rel) + leaky(edge_dot[edge_mat]) + m_bias
__global__ void g_kernel(const float* __restrict__ L, const float* __restrict__ men2rel,
                         const float* __restrict__ score_w, const float* __restrict__ edge_dot,
                         const int* __restrict__ edge_mat, const float* __restrict__ m_bias,
                         float* __restrict__ g) {
  int idx = blockIdx.x * 256 + threadIdx.x;  // 32768
  int b = idx >> 12, ij = idx & 4095;
  const float* p = men2rel + (size_t)b * 128 * 4096 + ij;
  float s = 0.0f;
  for (int c = 0; c < 128; ++c) s += score_w[c] * p[(size_t)c * 4096];
  float sc = LEAKY(s);
  float ed = edge_dot[edge_mat[idx]];
  ed = LEAKY(ed);
  g[idx] = L[idx] + sc + ed + m_bias[idx];
}

// K9: dual softmax of g in LDS, then
//   out_x[b] = y_lin[b](CINxE) @ G^T  + x[b]
//   out_y[b] = x_lin[b](CINxE) @ G2^T + y[b]
__global__ __launch_bounds__(256) void out_attn_kernel(const float* __restrict__ g,
                                                       const float* __restrict__ xlin,
                                                       const float* __restrict__ ylin,
                                                       const float* __restrict__ x,
                                                       const float* __restrict__ y,
                                                       float* __restrict__ out_x,
                                                       float* __restrict__ out_y) {
  extern __shared__ float smem[];
  float* Gr = smem;            // [64][66]  softmax over j
  float* Gc = smem + 4224;     // [64][66]  softmax over i, stored [n][m]
  float* Xs = smem + 8448;     // [128][65] x_lin
  float* Ys = smem + 16768;    // [128][65] y_lin
  float* gt = smem + 25088;    // [64][65]
  int b = blockIdx.x;
  int tid = threadIdx.x, wave = tid >> 5, lane = tid & 31;
  const float* gg = g + (size_t)b * 4096;
  for (int t = tid; t < 4096; t += 256) gt[(t >> 6) * 65 + (t & 63)] = gg[t];
  for (int t = tid; t < 8192; t += 256) {
    Xs[(t >> 6) * 65 + (t & 63)] = xlin[(size_t)b * 8192 + t];
    Ys[(t >> 6) * 65 + (t & 63)] = ylin[(size_t)b * 8192 + t];
  }
  __syncthreads();
  for (int r = wave; r < 64; r += 8) {
    float v0 = gt[r * 65 + lane], v1 = gt[r * 65 + lane + 32];
    float mx = wave_max32(fmaxf(v0, v1));
    float e0 = __expf(v0 - mx), e1 = __expf(v1 - mx);
    float inv = 1.0f / wave_sum32(e0 + e1);
    Gr[r * 66 + lane] = e0 * inv; Gr[r * 66 + lane + 32] = e1 * inv;
  }
  for (int n = wave; n < 64; n += 8) {
    float v0 = gt[lane * 65 + n], v1 = gt[(lane + 32) * 65 + n];
    float mx = wave_max32(fmaxf(v0, v1));
    float e0 = __expf(v0 - mx), e1 = __expf(v1 - mx);
    float inv = 1.0f / wave_sum32(e0 + e1);
    Gc[n * 66 + lane] = e0 * inv; Gc[n * 66 + lane + 32] = e1 * inv;
  }
  __syncthreads();
  int m0 = wave * 16, hi = lane >> 4, lo = lane & 15;
  const float* xg = x + (size_t)b * 8192;
  const float* yg = y + (size_t)b * 8192;
  float* ox = out_x + (size_t)b * 8192;
  float* oy = out_y + (size_t)b * 8192;
  {
    v8f acc[4];
#pragma unroll
    for (int nt = 0; nt < 4; ++nt)
#pragma unroll
      for (int r = 0; r < 8; ++r)
        acc[nt][r] = xg[(size_t)(m0 + r + 8 * hi) * 64 + nt * 16 + lo];
    for (int k0 = 0; k0 < 64; k0 += 4) {
      int kh = k0 + (hi << 1);
      v2f a; a.x = Ys[(m0 + lo) * 65 + kh]; a.y = Ys[(m0 + lo) * 65 + kh + 1];
#pragma unroll
      for (int nt = 0; nt < 4; ++nt) {
        v2f bb = lds_pair(Gr + (nt * 16 + lo) * 66 + kh);
        acc[nt] = wmma_f32(a, bb, acc[nt]);
      }
    }
#pragma unroll
    for (int nt = 0; nt < 4; ++nt) {
      int col = nt * 16 + lo;
#pragma unroll
      for (int r = 0; r < 8; ++r) ox[(size_t)(m0 + r + 8 * hi) * 64 + col] = acc[nt][r];
    }
  }
  {
    v8f acc[4];
#pragma unroll
    for (int nt = 0; nt < 4; ++nt)
#pragma unroll
      for (int r = 0; r < 8; ++r)
        acc[nt][r] = yg[(size_t)(m0 + r + 8 * hi) * 64 + nt * 16 + lo];
    for (int k0 = 0; k0 < 64; k0 += 4) {
      int kh = k0 + (hi << 1);
      v2f a; a.x = Xs[(m0 + lo) * 65 + kh]; a.y = Xs[(m0 + lo) * 65 + kh + 1];
#pragma unroll
      for (int nt = 0; nt < 4; ++nt) {
        v2f bb = lds_pair(Gc + (nt * 16 + lo) * 66 + kh);
        acc[nt] = wmma_f32(a, bb, acc[nt]);
      }
    }
#pragma unroll
    for (int nt = 0; nt < 4; ++nt) {
      int col = nt * 16 + lo;
#pragma unroll
      for (int r = 0; r < 8; ++r) oy[(size_t)(m0 + r + 8 * hi) * 64 + col] = acc[nt][r];
    }
  }
}

// ---------------------------------------------------------------------------
extern "C" void kernel_launch(void* const* d_in, const int* in_sizes, int n_in,
                              void* d_out, int out_size, void* d_ws, size_t ws_size,
                              hipStream_t stream) {
  const float* x        = (const float*)d_in[0];
  const float* y        = (const float*)d_in[1];
  const float* m_bias   = (const float*)d_in[2];
  const int*   edge_mat = (const int*)d_in[3];
  const float* p_bias   = (const float*)d_in[4];
  const int*   path_mat = (const int*)d_in[5];
  const float* pre_out  = (const float*)d_in[6];
  const float* xc1_w    = (const float*)d_in[7];
  const float* yc1_w    = (const float*)d_in[8];
  const float* xc2_w    = (const float*)d_in[9];
  const float* xc2_b    = (const float*)d_in[10];
  const float* yc2_w    = (const float*)d_in[11];
  const float* yc2_b    = (const float*)d_in[12];
  const float* pconv_w  = (const float*)d_in[13];
  const float* econv_w  = (const float*)d_in[14];
  const float* conv1_w  = (const float*)d_in[15];
  const float* conv1_b  = (const float*)d_in[16];
  const float* conv2_w  = (const float*)d_in[17];
  const float* conv2_b  = (const float*)d_in[18];
  const float* score_w  = (const float*)d_in[19];
  const float* xlin_w   = (const float*)d_in[20];
  const float* ylin_w   = (const float*)d_in[21];
  const float* edge_tab = (const float*)d_in[22];
  const float* path_tab = (const float*)d_in[23];

  float* out     = (float*)d_out;
  float* out_x   = out;            // [8,128,64]
  float* out_y   = out + 65536;    // [8,128,64]
  float* men2rel = out + 131072;   // [8,128,64,64]

  float* ws     = (float*)d_ws;
  float* pdot   = ws;                    // 32
  float* edot   = ws + 32;               // 32
  float* xf     = ws + 64;               // 65536
  float* yf     = xf + 65536;            // 65536
  float* xlin   = yf + 65536;            // 65536
  float* ylin   = xlin + 65536;          // 65536
  float* xself  = ylin + 65536;          // 512
  float* yoth   = xself + 512;           // 512
  float* Lb     = yoth + 512;            // 32768
  float* gbuf   = Lb + 32768;            // 32768
  float* w1t    = gbuf + 32768;          // 49152
  float* w2t    = w1t + 49152;           // 147456
  float* locbuf = w2t + 147456;          // 2097152
  float* xret   = locbuf + 2097152;      // 4194304
  float* yret   = xret + 4194304;        // 4194304
  float* h1     = yret + 4194304;        // 4194304

  const size_t SM_GEMM  = (size_t)(64 * 160 + 128 * 129) * 4;         // 107008
  const size_t SM_ATTN  = (size_t)(2 * 64 * 66 + 2 * 128 * 65 + 64 * 65) * 4;  // 116992
  const size_t SM_CONV1 = (size_t)(32 * 288 + 128 * 65) * 4;          // 70144
  const size_t SM_CONV2 = (size_t)(3 * 64 * 160 + 128 * 65) * 4;      // 156160

  prep_tables_kernel<<<1, 64, 0, stream>>>(path_tab, pconv_w, edge_tab, econv_w, pdot, edot);
  transpose_w1_kernel<<<192, 256, 0, stream>>>(conv1_w, w1t);
  prep_w2_kernel<<<576, 256, 0, stream>>>(conv2_w, w2t);

  gemm_wx_kernel<<<8, 256, SM_GEMM, stream>>>(xc1_w, x, xf);
  gemm_wx_kernel<<<8, 256, SM_GEMM, stream>>>(yc1_w, y, yf);
  gemm_wx_kernel<<<8, 256, SM_GEMM, stream>>>(xlin_w, x, xlin);
  gemm_wx_kernel<<<8, 256, SM_GEMM, stream>>>(ylin_w, y, ylin);

  self_kernel<<<4, 256, 0, stream>>>(xf, yf, xc2_w, xc2_b, yc2_w, yc2_b, xself, yoth);
  logist_kernel<<<128, 256, 0, stream>>>(xself, yoth, Lb);
  local_kernel<<<8192, 256, 0, stream>>>(Lb, pdot, path_mat, p_bias, locbuf);

  attn_ret_kernel<<<512, 256, SM_ATTN, stream>>>(locbuf, xf, yf, xret, yret);
  conv1_kernel<<<256, 256, SM_CONV1, stream>>>(xret, yret, pre_out, w1t, conv1_b, h1);
  conv2_kernel<<<512, 256, SM_CONV2, stream>>>(h1, w2t, conv2_b, men2rel);

  g_kernel<<<128, 256, 0, stream>>>(Lb, men2rel, score_w, edot, edge_mat, m_bias, gbuf);
  out_attn_kernel<<<8, 256, SM_ATTN, stream>>>(gbuf, xlin, ylin, x, y, out_x, out_y);
}